// PointNet2MSGCls_53102975648412
// MI455X (gfx1250) — compile-verified
//
#include <hip/hip_runtime.h>
#include <hip/hip_bf16.h>
#include <stdint.h>

// ---------------------------------------------------------------------------
// PointNet++ MSG classifier forward, CDNA5 (gfx1250), wave32.
// GEMMs: V_WMMA_F32_16X16X4_F32, one wave computes a 16x64 output block
// (4 sub-tiles) so each K-step is 1 A-load + 4 B-loads -> 4 WMMAs.
// No predication in the hot loop: M % 16 == 0 always; OOB weight rows are
// clamped (their products land in never-stored accumulator columns).
// Workspace requirement: ~960 MB (bufA 1M*96 f32, bufB 1M*128 f32 + misc).
// ---------------------------------------------------------------------------

#define B_  16
#define N_  4096
#define S0_ 512
#define S1_ 128

typedef float v2f __attribute__((ext_vector_type(2)));
typedef float v8f __attribute__((ext_vector_type(8)));

// ------------------------- transpose (B,3,N) -> (B,N,3) --------------------
__global__ void k_transpose(const float* __restrict__ pts, float* __restrict__ xyz) {
  int i = blockIdx.x * blockDim.x + threadIdx.x;
  if (i >= B_ * N_) return;
  int b = i / N_, n = i % N_;
  const float* p = pts + (size_t)b * 3 * N_;
  float* o = xyz + (size_t)i * 3;
  o[0] = p[0 * N_ + n];
  o[1] = p[1 * N_ + n];
  o[2] = p[2 * N_ + n];
}

// ------------------------- farthest point sampling -------------------------
__global__ void k_fps(const float* __restrict__ xyz, int* __restrict__ out,
                      int N, int S) {
  __shared__ float dist[N_];
  __shared__ float rv[256];
  __shared__ int   ri[256];
  int b = blockIdx.x, tid = threadIdx.x;
  const float* p = xyz + (size_t)b * N * 3;
  for (int j = tid; j < N; j += 256) dist[j] = 3.4e38f;
  __syncthreads();
  int far = 0;
  for (int s = 0; s < S; ++s) {
    if (tid == 0) out[b * S + s] = far;
    float cx = p[far * 3 + 0], cy = p[far * 3 + 1], cz = p[far * 3 + 2];
    float bv = -1.0f; int bi = 0;
    for (int j = tid; j < N; j += 256) {
      float dx = p[j * 3 + 0] - cx;
      float dy = p[j * 3 + 1] - cy;
      float dz = p[j * 3 + 2] - cz;
      float d = fminf(dist[j], dx * dx + dy * dy + dz * dz);
      dist[j] = d;
      if (d > bv) { bv = d; bi = j; }   // strictly-greater keeps lowest index
    }
    rv[tid] = bv; ri[tid] = bi;
    __syncthreads();
    for (int off = 128; off > 0; off >>= 1) {
      if (tid < off) {
        float ov = rv[tid + off]; int oi = ri[tid + off];
        if (ov > rv[tid] || (ov == rv[tid] && oi < ri[tid])) { rv[tid] = ov; ri[tid] = oi; }
      }
      __syncthreads();
    }
    far = ri[0];
    __syncthreads();
  }
}

// gather centroids: out[b,s,:] = xyz[b, idx[b,s], :]
__global__ void k_gather_xyz(const float* __restrict__ xyz, const int* __restrict__ idx,
                             float* __restrict__ out, int N, int S) {
  int i = blockIdx.x * blockDim.x + threadIdx.x;
  if (i >= B_ * S) return;
  int b = i / S;
  int j = idx[i];
  const float* p = xyz + ((size_t)b * N + j) * 3;
  out[i * 3 + 0] = p[0];
  out[i * 3 + 1] = p[1];
  out[i * 3 + 2] = p[2];
}

// ------------------------- ball query --------------------------------------
__global__ void k_ball(const float* __restrict__ xyz, const float* __restrict__ nxyz,
                       int* __restrict__ nidx, int N, int S, int K, float r2) {
  int i = blockIdx.x * blockDim.x + threadIdx.x;
  if (i >= B_ * S) return;
  int b = i / S;
  const float* p = xyz + (size_t)b * N * 3;
  float qx = nxyz[i * 3 + 0], qy = nxyz[i * 3 + 1], qz = nxyz[i * 3 + 2];
  int* o = nidx + (size_t)i * K;
  int cnt = 0, first = 0;
  for (int j = 0; j < N && cnt < K; ++j) {
    float dx = p[j * 3 + 0] - qx;
    float dy = p[j * 3 + 1] - qy;
    float dz = p[j * 3 + 2] - qz;
    if (dx * dx + dy * dy + dz * dz < r2) {
      if (cnt == 0) first = j;
      o[cnt++] = j;
    }
  }
  for (; cnt < K; ++cnt) o[cnt] = first;
}

// group: X[row, 0:3] = xyz[j]-center ; X[row, 3:3+Cf] = feat[j]
__global__ void k_group(const float* __restrict__ xyz, const float* __restrict__ nxyz,
                        const float* __restrict__ feat, const int* __restrict__ nidx,
                        float* __restrict__ X, int N, int S, int K, int Cf) {
  long long total = (long long)B_ * S * K;
  long long stride = (long long)gridDim.x * blockDim.x;
  for (long long r = blockIdx.x * (long long)blockDim.x + threadIdx.x; r < total; r += stride) {
    long long bs = r / K;
    int b = (int)(bs / S);
    int j = nidx[r];
    const float* p = xyz + ((size_t)b * N + j) * 3;
    const float* q = nxyz + (size_t)bs * 3;
    float* x = X + (size_t)r * (3 + Cf);
    x[0] = p[0] - q[0];
    x[1] = p[1] - q[1];
    x[2] = p[2] - q[2];
    if (Cf > 0) {
      const float* f = feat + ((size_t)b * N + j) * Cf;
      for (int c = 0; c < Cf; ++c) x[3 + c] = f[c];
    }
  }
}

// concat for sa2: X[r, 0:3] = new_xyz1[r]; X[r, 3:643] = feat1[r]
__global__ void k_concat_sa2(const float* __restrict__ nx1, const float* __restrict__ feat1,
                             float* __restrict__ X) {
  int r = blockIdx.x * blockDim.x + threadIdx.x;
  if (r >= B_ * S1_) return;
  float* x = X + (size_t)r * 643;
  x[0] = nx1[r * 3 + 0];
  x[1] = nx1[r * 3 + 1];
  x[2] = nx1[r * 3 + 2];
  const float* f = feat1 + (size_t)r * 640;
  for (int c = 0; c < 640; ++c) x[3 + c] = f[c];
}

// ------------------------- WMMA fp32 GEMM ----------------------------------
// Y[M,Cout] = X[M,Cin] * W[Cout,Cin]^T (+bias).  Requires M % 16 == 0.
// One wave -> 16(M) x 64(N) output block = 4 accumulators.
// A 16x4 layout: vgpr0: lanes0-15 K=k0, lanes16-31 K=k0+2; vgpr1: k0+1/k0+3.
// B 4x16 symmetric (N across lanes). C/D: vgpr v -> (M = v + 8*half, N = lane%16).
// VEC=true: Cin even -> rows 8B-aligned -> v2f (b64) loads.
template <bool VEC>
__global__ void k_gemm_wmma(const float* __restrict__ X, const float* __restrict__ W,
                            const float* __restrict__ bias, float* __restrict__ Y,
                            int M, int Cin, int Cout) {
  const int gw   = (blockIdx.x * blockDim.x + threadIdx.x) >> 5;
  const int nw   = (gridDim.x * blockDim.x) >> 5;
  const int lane = threadIdx.x & 31;
  const int half = lane >> 4;
  const int l16  = lane & 15;
  const int tm   = M >> 4;               // M is a multiple of 16
  const int tnb  = (Cout + 63) >> 6;     // 64-wide N blocks
  const int tcnt = tm * tnb;
  const int kfull = Cin & ~3;
  const int kh = half << 1;
  for (int t = gw; t < tcnt; t += nw) {
    const int m0 = (t % tm) << 4;        // m fastest: waves stream A together
    const int n0 = (t / tm) << 6;
    const float* xr = X + (size_t)(m0 + l16) * Cin;
    const float* wr[4];
#pragma unroll
    for (int j = 0; j < 4; ++j) {
      int rb = n0 + j * 16 + l16;
      if (rb > Cout - 1) rb = Cout - 1;  // clamp: results masked at store
      wr[j] = W + (size_t)rb * Cin;
    }
    const v8f vzero = {0.f, 0.f, 0.f, 0.f, 0.f, 0.f, 0.f, 0.f};
    v8f acc[4] = {vzero, vzero, vzero, vzero};
#pragma unroll 2
    for (int k0 = 0; k0 < kfull; k0 += 4) {
      const int k = k0 + kh;
      v2f a, b[4];
      if constexpr (VEC) {
        a = *(const v2f*)(xr + k);
#pragma unroll
        for (int j = 0; j < 4; ++j) b[j] = *(const v2f*)(wr[j] + k);
      } else {
        a.x = xr[k]; a.y = xr[k + 1];
#pragma unroll
        for (int j = 0; j < 4; ++j) { b[j].x = wr[j][k]; b[j].y = wr[j][k + 1]; }
      }
#pragma unroll
      for (int j = 0; j < 4; ++j)
        acc[j] = __builtin_amdgcn_wmma_f32_16x16x4_f32(false, a, false, b[j],
                                                       (short)0, acc[j], false, false);
    }
    if (kfull < Cin) {                   // K tail (Cin % 4 != 0)
      const int k = kfull + kh;
      v2f a, b[4];
      a.x = (k     < Cin) ? xr[k]     : 0.f;
      a.y = (k + 1 < Cin) ? xr[k + 1] : 0.f;
#pragma unroll
      for (int j = 0; j < 4; ++j) {
        b[j].x = (k     < Cin) ? wr[j][k]     : 0.f;
        b[j].y = (k + 1 < Cin) ? wr[j][k + 1] : 0.f;
      }
#pragma unroll
      for (int j = 0; j < 4; ++j)
        acc[j] = __builtin_amdgcn_wmma_f32_16x16x4_f32(false, a, false, b[j],
                                                       (short)0, acc[j], false, false);
    }
#pragma unroll
    for (int j = 0; j < 4; ++j) {
      const int c = n0 + j * 16 + l16;
      if (c < Cout) {
        const float bv = bias ? bias[c] : 0.f;
        const size_t base = (size_t)(m0 + (half << 3)) * Cout + c;
#pragma unroll
        for (int v = 0; v < 8; ++v)
          Y[base + (size_t)v * Cout] = acc[j][v] + bv;
      }
    }
  }
}

// ------------------------- BatchNorm ---------------------------------------
__global__ void k_bn_stats(const float* __restrict__ Y, float* __restrict__ mean,
                           float* __restrict__ var, long long M, int C) {
  int c = blockIdx.x;
  int tid = threadIdx.x;
  float s = 0.f, s2 = 0.f;
  for (long long r = tid; r < M; r += blockDim.x) {
    float v = Y[(size_t)r * C + c];
    s += v; s2 += v * v;
  }
  __shared__ float sh[256], sh2[256];
  sh[tid] = s; sh2[tid] = s2;
  __syncthreads();
  for (int off = 128; off > 0; off >>= 1) {
    if (tid < off) { sh[tid] += sh[tid + off]; sh2[tid] += sh2[tid + off]; }
    __syncthreads();
  }
  if (tid == 0) {
    float m = sh[0] / (float)M;
    mean[c] = m;
    var[c]  = sh2[0] / (float)M - m * m;
  }
}

__global__ void k_bn_apply_relu(float* __restrict__ Y, const float* __restrict__ mean,
                                const float* __restrict__ var, const float* __restrict__ g,
                                const float* __restrict__ bb, long long M, int C) {
  long long total = M * (long long)C;
  long long stride = (long long)gridDim.x * blockDim.x;
  for (long long i = blockIdx.x * (long long)blockDim.x + threadIdx.x; i < total; i += stride) {
    int c = (int)(i % C);
    float v = Y[i];
    v = g[c] * (v - mean[c]) * rsqrtf(var[c] + 1e-5f) + bb[c];
    Y[i] = v > 0.f ? v : 0.f;
  }
}

// ------------------------- max pool over K ----------------------------------
__global__ void k_maxpool(const float* __restrict__ H, float* __restrict__ out,
                          int BS, int K, int C, int Ctot, int off) {
  long long i = blockIdx.x * (long long)blockDim.x + threadIdx.x;
  if (i >= (long long)BS * C) return;
  long long bs = i / C;
  int c = (int)(i % C);
  const float* h = H + (size_t)bs * K * C + c;
  float m = -3.4e38f;
  for (int k = 0; k < K; ++k) m = fmaxf(m, h[(size_t)k * C]);
  out[(size_t)bs * Ctot + off + c] = m;
}

__global__ void k_zero(float* __restrict__ out, long long n) {
  long long i = blockIdx.x * (long long)blockDim.x + threadIdx.x;
  if (i < n) out[i] = 0.f;
}

// ===========================================================================
// Host orchestration
// ===========================================================================
struct Lyr { const float* W; const float* b; const float* g; };

static inline int cdiv_i(long long a, long long b) { return (int)((a + b - 1) / b); }

static void run_gemm(const float* X, const float* W, const float* bias, float* Y,
                     long long M, int Cin, int Cout, hipStream_t st) {
  long long tcnt = (M >> 4) * (long long)((Cout + 63) >> 6);
  int blocks = (int)((tcnt + 7) / 8);
  if (blocks > 8192) blocks = 8192;
  if (blocks < 1) blocks = 1;
  if ((Cin & 1) == 0)
    k_gemm_wmma<true><<<blocks, 256, 0, st>>>(X, W, bias, Y, (int)M, Cin, Cout);
  else
    k_gemm_wmma<false><<<blocks, 256, 0, st>>>(X, W, bias, Y, (int)M, Cin, Cout);
}

static void run_layer(const float* X, const Lyr& L, float* Y, long long M,
                      int Cin, int Cout, float* mean, float* var, hipStream_t st) {
  run_gemm(X, L.W, nullptr, Y, M, Cin, Cout, st);
  k_bn_stats<<<Cout, 256, 0, st>>>(Y, mean, var, M, Cout);
  long long tot = M * (long long)Cout;
  int ab = cdiv_i(tot, 256); if (ab > 4096) ab = 4096;
  k_bn_apply_relu<<<ab, 256, 0, st>>>(Y, mean, var, L.g, L.b, M, Cout);
}

extern "C" void kernel_launch(void* const* d_in, const int* in_sizes, int n_in,
                              void* d_out, int out_size, void* d_ws, size_t ws_size,
                              hipStream_t stream) {
  (void)in_sizes; (void)n_in; (void)ws_size;
  const float* points = (const float*)d_in[0];

  // Param flattening order: pytree sorted keys: cls, fc, sa0, sa1, sa2;
  // per-layer dict {'W','g','b'} sorted -> W, b, g.
  int pi = 1;
  const float* clsW = (const float*)d_in[pi++];
  const float* clsb = (const float*)d_in[pi++];
  Lyr fc[2];
  for (int l = 0; l < 2; ++l) {
    fc[l].W = (const float*)d_in[pi++];
    fc[l].b = (const float*)d_in[pi++];
    fc[l].g = (const float*)d_in[pi++];
  }
  Lyr sa0[3][3], sa1[3][3], sa2[3];
  for (int br = 0; br < 3; ++br)
    for (int l = 0; l < 3; ++l) {
      sa0[br][l].W = (const float*)d_in[pi++];
      sa0[br][l].b = (const float*)d_in[pi++];
      sa0[br][l].g = (const float*)d_in[pi++];
    }
  for (int br = 0; br < 3; ++br)
    for (int l = 0; l < 3; ++l) {
      sa1[br][l].W = (const float*)d_in[pi++];
      sa1[br][l].b = (const float*)d_in[pi++];
      sa1[br][l].g = (const float*)d_in[pi++];
    }
  for (int l = 0; l < 3; ++l) {
    sa2[l].W = (const float*)d_in[pi++];
    sa2[l].b = (const float*)d_in[pi++];
    sa2[l].g = (const float*)d_in[pi++];
  }

  // Workspace layout (all chunks are multiples of 16 bytes -> alignment holds)
  char* w = (char*)d_ws;
  float* xyz   = (float*)w; w += sizeof(float) * (size_t)B_ * N_ * 3;
  float* nx0   = (float*)w; w += sizeof(float) * (size_t)B_ * S0_ * 3;
  float* nx1   = (float*)w; w += sizeof(float) * (size_t)B_ * S1_ * 3;
  float* feat0 = (float*)w; w += sizeof(float) * (size_t)B_ * S0_ * 320;
  float* feat1 = (float*)w; w += sizeof(float) * (size_t)B_ * S1_ * 640;
  float* meanb = (float*)w; w += sizeof(float) * 1024;
  float* varb  = (float*)w; w += sizeof(float) * 1024;
  int*   fpsi  = (int*)w;   w += sizeof(int) * (size_t)B_ * S0_;
  int*   nidx  = (int*)w;   w += sizeof(int) * (size_t)B_ * S0_ * 128;
  float* bufA  = (float*)w; w += sizeof(float) * (size_t)1048576 * 96;   // 384 MB
  float* bufB  = (float*)w; w += sizeof(float) * (size_t)1048576 * 128;  // 512 MB

  float* outf = (float*)d_out;

  // Zero all outputs first (logits overwritten later; key_idx stays 0).
  k_zero<<<cdiv_i(out_size, 256), 256, 0, stream>>>(outf, (long long)out_size);

  // (B,3,N) -> (B,N,3)
  k_transpose<<<cdiv_i((long long)B_ * N_, 256), 256, 0, stream>>>(points, xyz);

  const int   sa0K[3]  = {16, 32, 128};
  const float sa0R[3]  = {0.1f, 0.2f, 0.4f};
  const int   sa0D[3][4] = {{3,32,32,64},{3,64,64,128},{3,64,96,128}};
  const int   sa0Off[3] = {0, 64, 192};
  const int   sa1K[3]  = {32, 64, 128};
  const float sa1R[3]  = {0.2f, 0.4f, 0.8f};
  const int   sa1D[3][4] = {{323,64,64,128},{323,128,128,256},{323,128,128,256}};
  const int   sa1Off[3] = {0, 128, 384};

  // ---------------- stage 0 ----------------
  k_fps<<<B_, 256, 0, stream>>>(xyz, fpsi, N_, S0_);
  k_gather_xyz<<<cdiv_i((long long)B_ * S0_, 256), 256, 0, stream>>>(xyz, fpsi, nx0, N_, S0_);
  for (int br = 0; br < 3; ++br) {
    int K = sa0K[br];
    float r2 = sa0R[br] * sa0R[br];
    long long M = (long long)B_ * S0_ * K;
    k_ball<<<cdiv_i((long long)B_ * S0_, 128), 128, 0, stream>>>(xyz, nx0, nidx, N_, S0_, K, r2);
    int gb = cdiv_i(M, 256); if (gb > 4096) gb = 4096;
    k_group<<<gb, 256, 0, stream>>>(xyz, nx0, nullptr, nidx, bufA, N_, S0_, K, 0);
    const float* x = bufA;
    for (int l = 0; l < 3; ++l) {
      float* y = (l & 1) ? bufA : bufB;
      run_layer(x, sa0[br][l], y, M, sa0D[br][l], sa0D[br][l + 1], meanb, varb, stream);
      x = y;
    }
    int C = sa0D[br][3];
    k_maxpool<<<cdiv_i((long long)B_ * S0_ * C, 256), 256, 0, stream>>>(
        x, feat0, B_ * S0_, K, C, 320, sa0Off[br]);
  }

  // ---------------- stage 1 ----------------
  k_fps<<<B_, 256, 0, stream>>>(nx0, fpsi, S0_, S1_);
  k_gather_xyz<<<cdiv_i((long long)B_ * S1_, 256), 256, 0, stream>>>(nx0, fpsi, nx1, S0_, S1_);
  for (int br = 0; br < 3; ++br) {
    int K = sa1K[br];
    float r2 = sa1R[br] * sa1R[br];
    long long M = (long long)B_ * S1_ * K;
    k_ball<<<cdiv_i((long long)B_ * S1_, 128), 128, 0, stream>>>(nx0, nx1, nidx, S0_, S1_, K, r2);
    int gb = cdiv_i(M, 256); if (gb > 4096) gb = 4096;
    k_group<<<gb, 256, 0, stream>>>(nx0, nx1, feat0, nidx, bufA, S0_, S1_, K, 320);
    const float* x = bufA;
    for (int l = 0; l < 3; ++l) {
      float* y = (l & 1) ? bufA : bufB;
      run_layer(x, sa1[br][l], y, M, sa1D[br][l], sa1D[br][l + 1], meanb, varb, stream);
      x = y;
    }
    int C = sa1D[br][3];
    k_maxpool<<<cdiv_i((long long)B_ * S1_ * C, 256), 256, 0, stream>>>(
        x, feat1, B_ * S1_, K, C, 640, sa1Off[br]);
  }

  // ---------------- stage 2 (global) ----------------
  k_concat_sa2<<<cdiv_i((long long)B_ * S1_, 128), 128, 0, stream>>>(nx1, feat1, bufA);
  {
    const int d[4] = {643, 256, 512, 1024};
    long long M = (long long)B_ * S1_;
    const float* x = bufA;
    for (int l = 0; l < 3; ++l) {
      float* y = (l & 1) ? bufA : bufB;
      run_layer(x, sa2[l], y, M, d[l], d[l + 1], meanb, varb, stream);
      x = y;
    }
    // max over the 128 points -> (B,1024) into bufA
    k_maxpool<<<cdiv_i((long long)B_ * 1024, 256), 256, 0, stream>>>(
        x, bufA, B_, S1_, 1024, 1024, 0);
  }

  // ---------------- FC head ----------------
  run_layer(bufA, fc[0], bufB, B_, 1024, 512, meanb, varb, stream);
  run_layer(bufB, fc[1], bufA, B_, 512, 256, meanb, varb, stream);

  // classifier: logits -> d_out[0:640]; key_idx region stays zero.
  run_gemm(bufA, clsW, clsb, outf, B_, 256, 40, stream);
}